// FeatureFetcher_420906795175
// MI455X (gfx1250) — compile-verified
//
#include <hip/hip_runtime.h>

// FeatureFetcher: project 3D points through per-view cameras, bilinearly
// sample a [B,V,C,H,W] feature map, emit [B,V,C,N].
//
// MI455X strategy: this op is pure gather (no matmul -> no WMMA). The win is
// L2 locality: gridDim.z = channel-groups so that concurrently-launched
// blocks (x-fastest dispatch order) all gather from the SAME few 1.25MB
// channel planes, which stay resident in the 192MB L2. Feature map is then
// read from HBM ~once (335MB) instead of per-gather (multi-GB).
// Output is streamed with nontemporal (TH=NT) stores so the 205MB of writes
// do not evict the resident planes.

#define FF_V 4
#define FF_C 32
#define FF_H 512
#define FF_W 640
#define FF_PLANE (FF_H * FF_W)
#define FF_CG 4              // channels per thread (projection amortization)

__launch_bounds__(256)
__global__ void feature_fetch_kernel(const float* __restrict__ fm,   // [BV,C,H,W]
                                     const float* __restrict__ pts,  // [B,3,N]
                                     const float* __restrict__ Km,   // [BV,3,3]
                                     const float* __restrict__ Em,   // [BV,3,4]
                                     float* __restrict__ out,        // [BV,C,N]
                                     int N) {
  const int n  = blockIdx.x * blockDim.x + threadIdx.x;
  const int bv = blockIdx.y;              // uniform -> scalar loads
  const int c0 = blockIdx.z * FF_CG;      // first channel of this group
  if (n >= N) return;
  const int b = bv / FF_V;

  // --- point (coalesced b32 loads; hot in L2 across channel groups) ---
  const float px = pts[(size_t)(b * 3 + 0) * N + n];
  const float py = pts[(size_t)(b * 3 + 1) * N + n];
  const float pz = pts[(size_t)(b * 3 + 2) * N + n];

  // --- uniform camera matrices (index depends only on blockIdx.y -> SGPR) ---
  const float* __restrict__ E = Em + bv * 12;   // row-major 3x4
  const float* __restrict__ K = Km + bv * 9;    // row-major 3x3

  // tp = R*p + t
  const float tx = E[0] * px + E[1] * py + E[2]  * pz + E[3];
  const float ty = E[4] * px + E[5] * py + E[6]  * pz + E[7];
  const float tz = E[8] * px + E[9] * py + E[10] * pz + E[11];

  const bool  z_zero = (tz == 0.0f);
  const float zs = z_zero ? 1.0f : tz;
  const float nx = tx / zs;
  const float ny = ty / zs;

  // uv = K * [nx, ny, 1]
  float u = K[0] * nx + K[1] * ny + K[2];
  float v = K[3] * nx + K[4] * ny + K[5];
  if (z_zero) { u = 2.0f * (float)FF_W; v = 2.0f * (float)FF_H; }

  // grid_sample(align_corners=True): ix = (gx+1)*0.5*(W-1) == u - 0.5
  const float ix = u - 0.5f;
  const float iy = v - 0.5f;

  const float x0f = floorf(ix), y0f = floorf(iy);
  const float x1f = x0f + 1.0f, y1f = y0f + 1.0f;
  const float wx1 = ix - x0f, wx0 = 1.0f - wx1;
  const float wy1 = iy - y0f, wy0 = 1.0f - wy1;

  const float Wm1 = (float)(FF_W - 1), Hm1 = (float)(FF_H - 1);
  const bool vx0 = (x0f >= 0.0f) & (x0f <= Wm1);
  const bool vx1 = (x1f >= 0.0f) & (x1f <= Wm1);
  const bool vy0 = (y0f >= 0.0f) & (y0f <= Hm1);
  const bool vy1 = (y1f >= 0.0f) & (y1f <= Hm1);

  // zeros-padding folded into weights (matches reference: clip index, w*valid)
  const float w00 = (vx0 & vy0) ? wx0 * wy0 : 0.0f;
  const float w10 = (vx1 & vy0) ? wx1 * wy0 : 0.0f;
  const float w01 = (vx0 & vy1) ? wx0 * wy1 : 0.0f;
  const float w11 = (vx1 & vy1) ? wx1 * wy1 : 0.0f;

  const int xc0 = (int)fminf(fmaxf(x0f, 0.0f), Wm1);
  const int xc1 = (int)fminf(fmaxf(x1f, 0.0f), Wm1);
  const int yc0 = (int)fminf(fmaxf(y0f, 0.0f), Hm1);
  const int yc1 = (int)fminf(fmaxf(y1f, 0.0f), Hm1);

  // Corner offsets into the first plane of this channel group.
  // Base pointer is uniform (SGPR pair); offsets are 32-bit VGPRs -> saddr-form
  // global_load_b32 with per-plane immediate/add stepping.
  const float* __restrict__ base = fm + ((size_t)bv * FF_C + c0) * FF_PLANE;
  const int i00 = yc0 * FF_W + xc0;
  const int i10 = yc0 * FF_W + xc1;
  const int i01 = yc1 * FF_W + xc0;
  const int i11 = yc1 * FF_W + xc1;

  float* __restrict__ o = out + ((size_t)bv * FF_C + c0) * N + n;

  // 16 independent gathers issued before first use; all hit the handful of
  // L2-resident planes shared by concurrently-running blocks.
  float f00[FF_CG], f10[FF_CG], f01[FF_CG], f11[FF_CG];
#pragma unroll
  for (int j = 0; j < FF_CG; ++j) {
    const int poff = j * FF_PLANE;
    f00[j] = base[i00 + poff];
    f10[j] = base[i10 + poff];
    f01[j] = base[i01 + poff];
    f11[j] = base[i11 + poff];
  }
#pragma unroll
  for (int j = 0; j < FF_CG; ++j) {
    const float r = f00[j] * w00 + f10[j] * w10 + f01[j] * w01 + f11[j] * w11;
    // Nontemporal (TH=NT) store: stream the output past L2 so it does not
    // evict the feature planes we are keeping resident.
    __builtin_nontemporal_store(r, o + (size_t)j * N);
  }
}

extern "C" void kernel_launch(void* const* d_in, const int* in_sizes, int n_in,
                              void* d_out, int out_size, void* d_ws, size_t ws_size,
                              hipStream_t stream) {
  const float* fm  = (const float*)d_in[0];  // [B,V,C,H,W]
  const float* pts = (const float*)d_in[1];  // [B,3,N]
  const float* Km  = (const float*)d_in[2];  // [B,V,3,3]
  const float* Em  = (const float*)d_in[3];  // [B,V,3,4]
  float* out = (float*)d_out;                // [B,V,C,N]

  const int BV = in_sizes[0] / (FF_C * FF_H * FF_W);  // B*V
  const int B  = BV / FF_V;
  const int N  = in_sizes[1] / (3 * B);

  dim3 block(256, 1, 1);
  dim3 grid((N + 255) / 256, BV, FF_C / FF_CG);
  feature_fetch_kernel<<<grid, block, 0, stream>>>(fm, pts, Km, Em, out, N);
}